// EdgeConv2dBlock_12051678232714
// MI455X (gfx1250) — compile-verified
//
#include <hip/hip_runtime.h>

#define B_    8
#define C_    64
#define N_    4096
#define OUT_  64
#define K_    20
#define SKIP_ 6
#define CK_   58        // channels used for knn (C - SKIP)
#define NEG_SLOPE 0.2f
#define BN_EPS    1e-5f

typedef __attribute__((ext_vector_type(2))) float v2f;
typedef __attribute__((ext_vector_type(8))) float v8f;

// D = A(16x4 f32) * B(4x16 f32) + C(16x16 f32)  -- V_WMMA_F32_16X16X4_F32
// args: (neg_a, A, neg_b, B, c_mod, C, reuse_a, reuse_b)
__device__ __forceinline__ v8f wmma_f32(v2f a, v2f b, v8f c) {
  return __builtin_amdgcn_wmma_f32_16x16x4_f32(false, a, false, b, (short)0, c,
                                               false, false);
}

// ---------------------------------------------------------------------------
// Kernel 0: zero-padded knn-channel copy: xpad[b][c][n] = x[b][6+c][n] (c<58)
// Removes all divergent guards from the KNN hot loop.
// ---------------------------------------------------------------------------
__global__ void pad_kernel(const float* __restrict__ x, float* __restrict__ xpad) {
  int t = blockIdx.x * 256 + threadIdx.x;   // t in [0, B*64*N)
  int n = t & (N_ - 1);
  int c = (t >> 12) & 63;
  int b = t >> 18;
  float v = 0.f;
  if (c < CK_) v = x[(size_t)b * C_ * N_ + (SKIP_ + c) * N_ + n];
  xpad[t] = v;
}

// ---------------------------------------------------------------------------
// Kernel 1: squared norms over knn channels: sq[b*N+n] = sum_c xpad^2
// ---------------------------------------------------------------------------
__global__ void sq_kernel(const float* __restrict__ xpad, float* __restrict__ sq) {
  int t = blockIdx.x * 256 + threadIdx.x;   // t in [0, B*N)
  int b = t >> 12;
  int n = t & (N_ - 1);
  const float* xb = xpad + (size_t)b * C_ * N_;
  float s = 0.f;
  #pragma unroll 8
  for (int c = 0; c < CK_; ++c) {
    float v = xb[c * N_ + n];
    s += v * v;
  }
  sq[t] = s;
}

// ---------------------------------------------------------------------------
// Kernel 2: KNN. One wave handles 32 rows (two 16x16 WMMA accumulators),
// loops over all 256 column tiles. pd = 2*inner - sq_row - sq_col.
// Each lane owns one row and keeps top-20 (value,index) in registers.
// ---------------------------------------------------------------------------
__global__ __launch_bounds__(32) void knn_kernel(const float* __restrict__ xpad,
                                                 const float* __restrict__ sq,
                                                 int* __restrict__ idx) {
  __shared__ __align__(16) float pds[32][16];
  const int b    = blockIdx.y;
  const int n0   = blockIdx.x * 32;
  const int lane = threadIdx.x;
  const int hi   = lane >> 4;        // 0 or 1 (ISA A/B/C lane-half split)
  const int lo   = lane & 15;
  const float* xp  = xpad + (size_t)b * C_ * N_;
  const float* sqb = sq + b * N_;

  // A tiles (rows n0..n0+15 and n0+16..n0+31), 16 k-steps of 4 channels.
  // 32-bit A layout: lanes 0-15 hold K={0,1} (V0,V1), lanes 16-31 K={2,3}.
  v2f a0[16], a1[16];
  #pragma unroll
  for (int kk = 0; kk < 16; ++kk) {
    int c0 = 4 * kk + 2 * hi;
    a0[kk].x = xp[c0 * N_ + n0 + lo];
    a0[kk].y = xp[(c0 + 1) * N_ + n0 + lo];
    a1[kk].x = xp[c0 * N_ + n0 + 16 + lo];
    a1[kk].y = xp[(c0 + 1) * N_ + n0 + 16 + lo];
  }
  // row norms for the 8 M-values each lane sees in the D layout
  float sqr0[8], sqr1[8];
  #pragma unroll
  for (int r = 0; r < 8; ++r) {
    sqr0[r] = sqb[n0 + r + 8 * hi];
    sqr1[r] = sqb[n0 + 16 + r + 8 * hi];
  }

  float topv[K_];
  int   topi[K_];
  #pragma unroll
  for (int j = 0; j < K_; ++j) { topv[j] = -3.4e38f; topi[j] = 0; }
  float minv = -3.4e38f;
  int   minp = 0;

  for (int m0 = 0; m0 < N_; m0 += 16) {
    // ---- load phase: all 16 B-fragments (unconditional, coalesced) ----
    v2f bt[16];
    #pragma unroll
    for (int kk = 0; kk < 16; ++kk) {
      int c0 = 4 * kk + 2 * hi;
      bt[kk].x = xp[c0 * N_ + m0 + lo];
      bt[kk].y = xp[(c0 + 1) * N_ + m0 + lo];
    }
    // ---- WMMA phase: dense chain, two independent accumulators ----
    v8f acc0 = {}; v8f acc1 = {};
    #pragma unroll
    for (int kk = 0; kk < 16; ++kk) {
      acc0 = wmma_f32(a0[kk], bt[kk], acc0);
      acc1 = wmma_f32(a1[kk], bt[kk], acc1);
    }
    float sqc = sqb[m0 + lo];
    #pragma unroll
    for (int r = 0; r < 8; ++r) {
      pds[r + 8 * hi][lo]      = 2.f * acc0[r] - sqr0[r] - sqc;
      pds[16 + r + 8 * hi][lo] = 2.f * acc1[r] - sqr1[r] - sqc;
    }
    __syncthreads();
    // selection: lane L owns local row L; running-min cutoff keeps the common
    // path at one compare, replacement path fully unrolled (register resident)
    const float4* rowp = (const float4*)&pds[lane][0];
    #pragma unroll
    for (int g = 0; g < 4; ++g) {
      float4 v4 = rowp[g];
      float vv[4] = {v4.x, v4.y, v4.z, v4.w};
      #pragma unroll
      for (int e = 0; e < 4; ++e) {
        float v = vv[e];
        if (v > minv) {
          int cand = m0 + 4 * g + e;
          #pragma unroll
          for (int q = 0; q < K_; ++q)
            if (q == minp) { topv[q] = v; topi[q] = cand; }
          minv = topv[0]; minp = 0;
          #pragma unroll
          for (int q = 1; q < K_; ++q)
            if (topv[q] < minv) { minv = topv[q]; minp = q; }
        }
      }
    }
    __syncthreads();
  }
  int* op = idx + (size_t)(b * N_ + n0 + lane) * K_;
  #pragma unroll
  for (int j = 0; j < K_; ++j) op[j] = topi[j];
}

// ---------------------------------------------------------------------------
// Kernel 3: P = X^T * W1^T, Q = X^T * (W2-W1)^T   (per batch, N x 64 each)
// One wave: 16-row tile, all 4 col-tiles for both P and Q (128 WMMAs).
// ---------------------------------------------------------------------------
__global__ __launch_bounds__(32) void pq_kernel(const float* __restrict__ x,
                                                const float* __restrict__ W,
                                                float* __restrict__ P,
                                                float* __restrict__ Q) {
  const int b    = blockIdx.y;
  const int n0   = blockIdx.x * 16;
  const int lane = threadIdx.x;
  const int hi   = lane >> 4;
  const int lo   = lane & 15;
  const float* xb = x + (size_t)b * C_ * N_;

  v2f a[16];
  #pragma unroll
  for (int kk = 0; kk < 16; ++kk) {
    int c0 = 4 * kk + 2 * hi;
    a[kk].x = xb[c0 * N_ + n0 + lo];
    a[kk].y = xb[(c0 + 1) * N_ + n0 + lo];
  }

  v8f accP[4], accQ[4];
  v8f z = {};
  #pragma unroll
  for (int t = 0; t < 4; ++t) { accP[t] = z; accQ[t] = z; }

  #pragma unroll
  for (int t = 0; t < 4; ++t) {
    int o = 16 * t + lo;                    // B-matrix column = output chan
    const float* wr = W + o * (2 * C_);
    v2f b1[16], b2[16];
    #pragma unroll
    for (int kk = 0; kk < 16; ++kk) {
      int c0 = 4 * kk + 2 * hi;
      float w1x = wr[c0], w1y = wr[c0 + 1];
      b1[kk].x = w1x;                 b1[kk].y = w1y;
      b2[kk].x = wr[C_ + c0] - w1x;   b2[kk].y = wr[C_ + c0 + 1] - w1y;
    }
    #pragma unroll
    for (int kk = 0; kk < 16; ++kk) {
      accP[t] = wmma_f32(a[kk], b1[kk], accP[t]);
      accQ[t] = wmma_f32(a[kk], b2[kk], accQ[t]);
    }
  }

  #pragma unroll
  for (int t = 0; t < 4; ++t) {
    #pragma unroll
    for (int r = 0; r < 8; ++r) {
      int n = n0 + r + 8 * hi;
      int o = 16 * t + lo;
      size_t off = (size_t)(b * N_ + n) * OUT_ + o;
      P[off] = accP[t][r];
      Q[off] = accQ[t][r];
    }
  }
}

// ---------------------------------------------------------------------------
// Kernel 4: per-channel sum / sumsq of y = P[idx] + Q over (B,N,K).
// Block (64 chan, 4 points): P gathers coalesced on tx.
// ---------------------------------------------------------------------------
__global__ void stats_kernel(const float* __restrict__ P,
                             const float* __restrict__ Q,
                             const int* __restrict__ idx,
                             float* __restrict__ stats) {
  __shared__ float ssum[OUT_], ssq[OUT_];
  int tx = threadIdx.x, ty = threadIdx.y;
  if (ty == 0) { ssum[tx] = 0.f; ssq[tx] = 0.f; }
  __syncthreads();
  int bn = blockIdx.x * 4 + ty;
  int b  = bn >> 12;
  const float* Pb = P + (size_t)b * N_ * OUT_;
  float q = Q[(size_t)bn * OUT_ + tx];
  const int* ib = idx + (size_t)bn * K_;
  float s = 0.f, s2 = 0.f;
  #pragma unroll
  for (int k = 0; k < K_; ++k) {
    int id = ib[k];
    float y = Pb[(size_t)id * OUT_ + tx] + q;
    s += y; s2 += y * y;
  }
  atomicAdd(&ssum[tx], s);
  atomicAdd(&ssq[tx], s2);
  __syncthreads();
  if (ty == 0) {
    atomicAdd(&stats[tx], ssum[tx]);
    atomicAdd(&stats[OUT_ + tx], ssq[tx]);
  }
}

// Kernel 5: fold BN into per-channel scale a and shift bsh.
__global__ void finalize_kernel(const float* __restrict__ gamma,
                                const float* __restrict__ beta,
                                float* __restrict__ stats) {
  int o = threadIdx.x;
  const float inv_cnt = 1.0f / (float)((long)B_ * N_ * K_);
  float mean = stats[o] * inv_cnt;
  float var  = stats[OUT_ + o] * inv_cnt - mean * mean;
  float a    = gamma[o] * rsqrtf(var + BN_EPS);
  stats[2 * OUT_ + o] = a;
  stats[3 * OUT_ + o] = beta[o] - mean * a;
}

// Kernel 6: out[b,o,n] = max_k leakyrelu(a*(P[b,idx,o]+Q[b,n,o]) + bsh)
__global__ void out_kernel(const float* __restrict__ P,
                           const float* __restrict__ Q,
                           const int* __restrict__ idx,
                           const float* __restrict__ stats,
                           float* __restrict__ out) {
  int tx = threadIdx.x, ty = threadIdx.y;
  int bn = blockIdx.x * 4 + ty;
  int b  = bn >> 12;
  int n  = bn & (N_ - 1);
  const float* Pb = P + (size_t)b * N_ * OUT_;
  float q  = Q[(size_t)bn * OUT_ + tx];
  float a  = stats[2 * OUT_ + tx];
  float bs = stats[3 * OUT_ + tx];
  const int* ib = idx + (size_t)bn * K_;
  float mx = -3.4e38f;
  #pragma unroll
  for (int k = 0; k < K_; ++k) {
    int id = ib[k];
    float y = a * (Pb[(size_t)id * OUT_ + tx] + q) + bs;
    y = (y >= 0.f) ? y : NEG_SLOPE * y;
    mx = fmaxf(mx, y);
  }
  out[(((size_t)b * OUT_ + tx) << 12) + n] = mx;
}

__global__ void zero_kernel(float* __restrict__ stats) {
  stats[threadIdx.x] = 0.f;
}

// ---------------------------------------------------------------------------
extern "C" void kernel_launch(void* const* d_in, const int* in_sizes, int n_in,
                              void* d_out, int out_size, void* d_ws, size_t ws_size,
                              hipStream_t stream) {
  const float* x     = (const float*)d_in[0];   // (B, C, N)
  const float* W     = (const float*)d_in[1];   // (OUT, 2C)
  const float* gamma = (const float*)d_in[2];   // (OUT,)
  const float* beta  = (const float*)d_in[3];   // (OUT,)
  float* out = (float*)d_out;                   // (B, OUT, N)

  // workspace layout (~28 MB)
  char* ws = (char*)d_ws;
  float* sq    = (float*)ws;                                   // B*N       (128 KB)
  float* xpad  = (float*)(ws + 131072);                        // B*64*N    (8 MB)
  float* P     = (float*)(ws + 131072 + 8388608);              // B*N*OUT   (8 MB)
  float* Q     = (float*)(ws + 131072 + 2 * 8388608);          // B*N*OUT   (8 MB)
  float* stats = (float*)(ws + 131072 + 3 * 8388608);          // 256 floats
  int*   idx   = (int*)  (ws + 131072 + 3 * 8388608 + 1024);   // B*N*K     (2.6 MB)

  zero_kernel<<<1, 256, 0, stream>>>(stats);
  pad_kernel<<<(B_ * C_ * N_) / 256, 256, 0, stream>>>(x, xpad);
  sq_kernel<<<(B_ * N_) / 256, 256, 0, stream>>>(xpad, sq);
  knn_kernel<<<dim3(N_ / 32, B_), 32, 0, stream>>>(xpad, sq, idx);
  pq_kernel<<<dim3(N_ / 16, B_), 32, 0, stream>>>(x, W, P, Q);
  stats_kernel<<<(B_ * N_) / 4, dim3(64, 4), 0, stream>>>(P, Q, idx, stats);
  finalize_kernel<<<1, 64, 0, stream>>>(gamma, beta, stats);
  out_kernel<<<(B_ * N_) / 4, dim3(64, 4), 0, stream>>>(P, Q, idx, stats, out);
}